// ResidualVQ_55473797595574
// MI455X (gfx1250) — compile-verified
//
#include <hip/hip_runtime.h>
#include <hip/hip_bf16.h>

typedef __attribute__((ext_vector_type(2))) float v2f;
typedef __attribute__((ext_vector_type(8))) float v8f;

#define B_  16
#define D_  512
#define T_  2000
#define Q_  8
#define CD_ 8
#define CS_ 1024
#define TT  16          // tokens per tile
#define NBLK (B_ * (T_ / TT))   // 2000 workgroups

// ---------------------------------------------------------------------------
// Deterministic parallel loss finalize: fixed-shape strided partials + LDS
// tree (same reduction order every replay -> bit-deterministic).
// loss[st] = 1.25 * sum(diff^2) / (B*CD*T)
// ---------------------------------------------------------------------------
__global__ void rvq_loss_final(const float* __restrict__ part,
                               float* __restrict__ out_loss) {
  __shared__ float red[256];
  const int st  = blockIdx.x;
  const int tid = threadIdx.x;
  float acc = 0.f;
  for (int k = tid; k < NBLK; k += 256) acc += part[st * NBLK + k];
  red[tid] = acc;
  __syncthreads();
  for (int w = 128; w > 0; w >>= 1) {
    if (tid < w) red[tid] += red[tid + w];
    __syncthreads();
  }
  if (tid == 0) out_loss[st] = red[0] * (1.25f / (float)(B_ * CD_ * T_));
}

// ---------------------------------------------------------------------------
// Fused ResidualVQ: one workgroup = 16 tokens of one batch, all 8 stages.
// 4 waves; wave w owns dims [128w, 128w+128). Residual kept in registers in
// the f32 WMMA B-operand layout (fully-unrolled constant indexing only).
// quantized_out recovered as x - residual_final.
// waves_per_eu(4): <=256 VGPRs/wave, 4 waves/SIMD32.
// All hot-loop global accesses use one base register + 24-bit imm offsets.
// ---------------------------------------------------------------------------
__launch_bounds__(128)
__attribute__((amdgpu_waves_per_eu(4)))
__global__ void rvq_kernel(const float* __restrict__ x,
                           const float* __restrict__ Wi,   // [Q][CD][D]
                           const float* __restrict__ bi,   // [Q][CD]
                           const float* __restrict__ Wo,   // [Q][D][CD]
                           const float* __restrict__ bo,   // [Q][D]
                           const float* __restrict__ cb,   // [Q][CS][CD]
                           float* __restrict__ out_quant,  // [B][D][T]
                           float* __restrict__ out_idx,    // [Q][B][T] (as float)
                           float* __restrict__ loss_part,  // [Q][NBLK]
                           float* __restrict__ out_q)      // [Q][B][D][T]
{
  __shared__ float s_cbn[CS_][CD_];     // normalized codebook rows
  __shared__ float s_cbnorm[CS_];       // row norms (raw = cbn * norm)
  __shared__ float s_ze[CD_][TT];
  __shared__ float s_enc[CD_][TT];
  __shared__ float s_zq[CD_][TT];
  __shared__ float s_tnorm[TT];
  __shared__ float s_zpart[4][8][32];   // per-wave z_e partial C tiles
  __shared__ float s_bv[4][32];         // per-wave argmax candidates
  __shared__ int   s_bix[4][32];
  __shared__ int   s_tokidx[TT];
  __shared__ float s_qstage[4][16][TT]; // per-wave q tile restage (C -> B layout)
  __shared__ float s_lred[128];

  const int tid  = threadIdx.x;
  const int lane = tid & 31;
  const int wv   = tid >> 5;
  const int j    = lane & 15;     // token column (N) held by this lane
  const int hi   = lane >> 4;     // half-wave select
  const int m    = j;             // A-operand row (M) held by this lane
  const int blk  = blockIdx.x;
  const int b    = blk / (T_ / TT);
  const int t0   = (blk % (T_ / TT)) * TT;

  // B-operand K placement per half-wave: VGPR0 holds K{0|1}, VGPR1 holds K{2|3}
  const int off0 = hi;
  const int off1 = 2 + hi;
  const int dbase = wv * 128;

  // row mask for the 8-row A operands (rows 8..15 must contribute zero)
  const float amask = (m < CD_) ? 1.f : 0.f;
  const int   mrow  = m & 7;      // clamped row -> always a valid address

  // ---- residual tile -> registers (WMMA B-operand layout) ----
  float Rreg[64];
  const float* xb = x + (size_t)b * D_ * T_ + t0 + j;
#pragma unroll
  for (int s = 0; s < 32; ++s) {
    Rreg[2 * s + 0] = xb[(size_t)(dbase + 4 * s + off0) * T_];
    Rreg[2 * s + 1] = xb[(size_t)(dbase + 4 * s + off1) * T_];
  }

  for (int st = 0; st < Q_; ++st) {
    // Prefetch next stage codebook toward L2 while we work on this one.
    if (st + 1 < Q_)
      __builtin_prefetch(cb + (size_t)(st + 1) * CS_ * CD_ + tid * 16, 0, 0);

    // ---- load + L2-normalize codebook into LDS (8 rows per thread) ----
    const float* cbs = cb + (size_t)st * CS_ * CD_;
    for (int r = tid; r < CS_; r += 128) {
      float v[8], ss = 0.f;
#pragma unroll
      for (int c = 0; c < 8; ++c) { v[c] = cbs[r * 8 + c]; ss += v[c] * v[c]; }
      float nr = sqrtf(ss);
      float inv = 1.f / fmaxf(nr, 1e-12f);
      s_cbnorm[r] = nr;
#pragma unroll
      for (int c = 0; c < 8; ++c) s_cbn[r][c] = v[c] * inv;
    }
    __syncthreads();

    // ---- down-projection: z_e = Wi @ R  (M=8 chan, N=16 tok, K=512/4waves) ----
    const float* wrow = Wi + (size_t)st * CD_ * D_ + (size_t)mrow * D_ + 2 * hi;
    v8f cz = {};
#pragma unroll
    for (int ks = 0; ks < 32; ++ks) {
      const int d0 = dbase + 4 * ks;
      v2f a = *(const v2f*)(wrow + d0);   // global_load_b64, imm offset
      a.x *= amask;
      a.y *= amask;
      v2f bv2;
      bv2.x = Rreg[2 * ks + 0];
      bv2.y = Rreg[2 * ks + 1];
      cz = __builtin_amdgcn_wmma_f32_16x16x4_f32(false, a, false, bv2,
                                                 (short)0, cz, false, false);
    }
#pragma unroll
    for (int r = 0; r < 8; ++r) s_zpart[wv][r][lane] = cz[r];
    __syncthreads();

    // reduce wave partials + bias (rows M<8 live on lanes 0..15, VGPR r = M)
    {
      const int c = tid >> 4, jj = tid & 15;
      float z = bi[st * CD_ + c]
              + s_zpart[0][c][jj] + s_zpart[1][c][jj]
              + s_zpart[2][c][jj] + s_zpart[3][c][jj];
      s_ze[c][jj] = z;
    }
    __syncthreads();

    // ---- per-token L2 normalize ----
    if (tid < TT) {
      float ss = 0.f;
#pragma unroll
      for (int c = 0; c < 8; ++c) { float z = s_ze[c][tid]; ss += z * z; }
      s_tnorm[tid] = 1.f / fmaxf(sqrtf(ss), 1e-12f);
    }
    __syncthreads();
    {
      const int c = tid >> 4, jj = tid & 15;
      s_enc[c][jj] = s_ze[c][jj] * s_tnorm[jj];
    }
    __syncthreads();

    // ---- cosine sim + argmax: 64 code tiles of 16, 2 WMMA each (K=8) ----
    v2f b0, b1;
    b0.x = s_enc[0 + hi][j];  b0.y = s_enc[2 + hi][j];   // K0..K3
    b1.x = s_enc[4 + hi][j];  b1.y = s_enc[6 + hi][j];   // K4..K7
    float bv = -1e30f; int bidx = 0;
#pragma unroll
    for (int i = 0; i < 16; ++i) {
      const int m0 = (wv + 4 * i) * 16;
      v2f a0 = *(const v2f*)(&s_cbn[m0 + m][2 * hi]);      // ds_load_b64
      v2f a1 = *(const v2f*)(&s_cbn[m0 + m][4 + 2 * hi]);
      v8f sc = {};
      sc = __builtin_amdgcn_wmma_f32_16x16x4_f32(false, a0, false, b0,
                                                 (short)0, sc, false, false);
      sc = __builtin_amdgcn_wmma_f32_16x16x4_f32(false, a1, false, b1,
                                                 (short)0, sc, false, false);
#pragma unroll
      for (int r = 0; r < 8; ++r) {
        float v = sc[r];
        int idx = m0 + r + 8 * hi;
        if (v > bv || (v == bv && idx < bidx)) { bv = v; bidx = idx; }
      }
    }
    s_bv[wv][lane]  = bv;
    s_bix[wv][lane] = bidx;
    __syncthreads();

    // reduce argmax across 4 waves x 2 half-waves; lookup raw codebook row
    if (tid < TT) {
      float best = -1e30f; int besti = 0;
#pragma unroll
      for (int w = 0; w < 4; ++w)
#pragma unroll
        for (int h = 0; h < 2; ++h) {
          float v = s_bv[w][tid + 16 * h];
          int  ix = s_bix[w][tid + 16 * h];
          if (v > best || (v == best && ix < besti)) { best = v; besti = ix; }
        }
      s_tokidx[tid] = besti;
      out_idx[((size_t)st * B_ + b) * T_ + t0 + tid] = (float)besti;
    }
    __syncthreads();
    {
      const int c = tid >> 4, jj = tid & 15;
      int ix = s_tokidx[jj];
      float zqv = s_cbn[ix][c] * s_cbnorm[ix];   // raw codebook value
      s_zq[c][jj] = zqv;
      float d = s_ze[c][jj] - zqv;
      s_lred[tid] = d * d;                       // commit == cb loss in value
    }
    __syncthreads();
    if (tid == 0) {
      float acc = 0.f;
      for (int k = 0; k < 128; ++k) acc += s_lred[k];
      loss_part[st * NBLK + blk] = acc;          // deterministic partial
    }

    // ---- up-projection: q = Wo @ zq + bo ; fused all_q / residual update ----
    // Single per-stage store base; (16*mt + r)*T fits the 24-bit imm offset.
    const float* Wos  = Wo + (size_t)st * D_ * CD_ + (size_t)(dbase + m) * 8 + 2 * hi;
    const float* bos  = bo + (size_t)st * D_;
    float* outq_base  = out_q + (((size_t)st * B_ + b) * D_ + (dbase + 8 * hi)) * T_
                              + t0 + j;

    v2f zb0, zb1;
    zb0.x = s_zq[0 + hi][j];  zb0.y = s_zq[2 + hi][j];
    zb1.x = s_zq[4 + hi][j];  zb1.y = s_zq[6 + hi][j];

#pragma unroll
    for (int mt = 0; mt < 8; ++mt) {
      // A pairs adjacent in memory -> global_load_b64 (base + mt*512B imm).
      v2f a0 = *(const v2f*)(Wos + mt * 128);
      v2f a1 = *(const v2f*)(Wos + mt * 128 + 4);
      v8f cq = {};
      cq = __builtin_amdgcn_wmma_f32_16x16x4_f32(false, a0, false, zb0,
                                                 (short)0, cq, false, false);
      cq = __builtin_amdgcn_wmma_f32_16x16x4_f32(false, a1, false, zb1,
                                                 (short)0, cq, false, false);
#pragma unroll
      for (int r = 0; r < 8; ++r) {
        const int drow = r + 8 * hi;
        float v = cq[r] + bos[dbase + 16 * mt + drow];
        outq_base[(size_t)(16 * mt + r) * T_] = v;   // one base + imm offsets
        s_qstage[wv][drow][j] = v;                   // restage C -> B layout
      }
      __builtin_amdgcn_wave_barrier();
#pragma unroll
      for (int s2 = 0; s2 < 4; ++s2) {
        const int sidx = 4 * mt + s2;                // constant after unroll
        Rreg[2 * sidx + 0] -= s_qstage[wv][4 * s2 + off0][j];
        Rreg[2 * sidx + 1] -= s_qstage[wv][4 * s2 + off1][j];
      }
      __builtin_amdgcn_wave_barrier();
    }
    __syncthreads();   // protect s_cbn / s_zq / s_lred before next stage
  }

  // ---- quantized_out = sum of stage q's = x - residual_final ----
  float* outq0 = out_quant + ((size_t)b * D_) * T_ + t0 + j;
#pragma unroll
  for (int s = 0; s < 32; ++s) {
    const size_t o0 = (size_t)(dbase + 4 * s + off0) * T_;
    const size_t o1 = (size_t)(dbase + 4 * s + off1) * T_;
    outq0[o0] = xb[o0] - Rreg[2 * s + 0];
    outq0[o1] = xb[o1] - Rreg[2 * s + 1];
  }
}

// ---------------------------------------------------------------------------
extern "C" void kernel_launch(void* const* d_in, const int* in_sizes, int n_in,
                              void* d_out, int out_size, void* d_ws, size_t ws_size,
                              hipStream_t stream) {
  const float* x  = (const float*)d_in[0];
  const float* Wi = (const float*)d_in[1];
  const float* bi = (const float*)d_in[2];
  const float* Wo = (const float*)d_in[3];
  const float* bo = (const float*)d_in[4];
  const float* cb = (const float*)d_in[5];

  float* out = (float*)d_out;
  // tuple layout: quantized_out | all_idx | all_loss | all_q
  float* out_quant = out;
  float* out_idx   = out + (size_t)B_ * D_ * T_;                       // 16,384,000
  float* out_loss  = out_idx + (size_t)Q_ * B_ * T_;                   // +256,000
  float* out_q     = out_loss + Q_;                                    // +8

  float* loss_part = (float*)d_ws;   // Q_ * NBLK floats = 64 KB

  rvq_kernel<<<NBLK, 128, 0, stream>>>(x, Wi, bi, Wo, bo, cb,
                                       out_quant, out_idx, loss_part, out_q);
  rvq_loss_final<<<Q_, 256, 0, stream>>>(loss_part, out_loss);
}